// HyperGraphConvolution_7404523618362
// MI455X (gfx1250) — compile-verified
//
#include <hip/hip_runtime.h>

// ---------------------------------------------------------------------------
// HyperGraphConvolution for MI455X (gfx1250, wave32)
//   support = X @ W           -> fp32 WMMA 16x16x4 (exact fp32 path)
//   out     = COO-SpMM(support) via global_atomic_add_f32 scatter
//   out     = relu(out) + bias
// Workspace: support_1 (20000*64 f32) followed by support_2 (400000*64 f32).
// ---------------------------------------------------------------------------

typedef __attribute__((ext_vector_type(2))) float v2f;
typedef __attribute__((ext_vector_type(8))) float v8f;

#define HGC_D 64

// -------- GEMM: S[R,64] = X[R,64] @ W[64,64] via V_WMMA_F32_16X16X4_F32 -----
// One wave computes one 16x16 output tile; 4 waves per block span N=64.
// K=64 is consumed in 16 WMMA steps of K=4.
__global__ __launch_bounds__(128) void hgc_gemm_xw_wmma(
    const float* __restrict__ X, const float* __restrict__ W,
    float* __restrict__ S, int R) {
  const int lane = threadIdx.x & 31;
  const int wave = threadIdx.x >> 5;     // 0..3 => N-tile
  const int m0   = blockIdx.x * 16;      // row tile base
  if (m0 >= R) return;
  const int n0 = wave * 16;
  const int ml = lane & 15;              // M (A) / N (B,C) within tile
  const int kh = (lane >> 4) * 2;        // lane-half K offset: 0 or 2

  v8f c = {0.f, 0.f, 0.f, 0.f, 0.f, 0.f, 0.f, 0.f};
  const float* Xrow = X + (size_t)(m0 + ml) * HGC_D;

#pragma unroll
  for (int kk = 0; kk < HGC_D; kk += 4) {
    v2f a, b;
    // A 16x4 fp32: lanes 0-15 -> K = kk,kk+1 ; lanes 16-31 -> K = kk+2,kk+3
    a.x = Xrow[kk + kh + 0];
    a.y = Xrow[kk + kh + 1];
    // B 4x16 fp32: row K striped across lanes within a VGPR (halves split K)
    b.x = W[(size_t)(kk + kh + 0) * HGC_D + n0 + ml];
    b.y = W[(size_t)(kk + kh + 1) * HGC_D + n0 + ml];
    c = __builtin_amdgcn_wmma_f32_16x16x4_f32(
        /*neg_a=*/false, a, /*neg_b=*/false, b,
        /*c_mod=*/(short)0, c, /*reuse_a=*/false, /*reuse_b=*/false);
  }

  // C/D 16x16 fp32: VGPR r -> row r (lanes 0-15) / row r+8 (lanes 16-31)
  const int rbase = m0 + (lane >> 4) * 8;
#pragma unroll
  for (int r = 0; r < 8; ++r) {
    S[(size_t)(rbase + r) * HGC_D + n0 + ml] = c[r];
  }
}

// -------- zero the accumulation target --------------------------------------
__global__ void hgc_zero(float* __restrict__ p, long long n) {
  long long i = (long long)blockIdx.x * blockDim.x + threadIdx.x;
  long long stride = (long long)gridDim.x * blockDim.x;
  for (; i < n; i += stride) p[i] = 0.0f;
}

// -------- COO SpMM scatter: out[row] += val * S[col] ------------------------
// 16 threads per nonzero; each thread handles 4 contiguous columns (float4
// gather from L2-resident support, 4 global f32 atomic adds scatter).
__global__ __launch_bounds__(256) void hgc_spmm_scatter(
    const int* __restrict__ rows, const int* __restrict__ cols,
    const float* __restrict__ vals, const float* __restrict__ S,
    float* __restrict__ out, int nnz) {
  long long tid = (long long)blockIdx.x * blockDim.x + threadIdx.x;
  long long e = tid >> 4;
  if (e >= nnz) return;
  const int q = (int)(tid & 15) * 4;
  const int r = rows[e];
  const int c = cols[e];
  const float v = vals[e];
  const float4 s4 = *(const float4*)(S + (size_t)c * HGC_D + q);
  float* o = out + (size_t)r * HGC_D + q;
  atomicAdd(o + 0, v * s4.x);
  atomicAdd(o + 1, v * s4.y);
  atomicAdd(o + 2, v * s4.z);
  atomicAdd(o + 3, v * s4.w);
}

// -------- epilogue: out = relu(out) + bias ----------------------------------
__global__ void hgc_relu_bias(float* __restrict__ out,
                              const float* __restrict__ bias, long long n) {
  long long i = (long long)blockIdx.x * blockDim.x + threadIdx.x;
  long long stride = (long long)gridDim.x * blockDim.x;
  for (; i < n; i += stride) {
    out[i] = fmaxf(out[i], 0.0f) + bias[i & (HGC_D - 1)];
  }
}

extern "C" void kernel_launch(void* const* d_in, const int* in_sizes, int n_in,
                              void* d_out, int out_size, void* d_ws, size_t ws_size,
                              hipStream_t stream) {
  const float* node_input = (const float*)d_in[0];
  const float* he_input   = (const float*)d_in[1];
  const int*   nrows      = (const int*)d_in[2];
  const int*   ncols      = (const int*)d_in[3];
  const float* nvals      = (const float*)d_in[4];
  const int*   hrows      = (const int*)d_in[5];
  const int*   hcols      = (const int*)d_in[6];
  const float* hvals      = (const float*)d_in[7];
  const float* weight     = (const float*)d_in[8];
  const float* bias       = (const float*)d_in[9];

  const int Rn    = in_sizes[0] / HGC_D;   // 20000 nodes
  const int Rh    = in_sizes[1] / HGC_D;   // 400000 hyperedges
  const int nnz_n = in_sizes[2];           // 640000
  const int nnz_h = in_sizes[5];           // 6400000

  float* S1 = (float*)d_ws;                       // support_1 [Rn, 64]
  float* S2 = S1 + (size_t)Rn * HGC_D;            // support_2 [Rh, 64]
  float* out1 = (float*)d_out;                    // [Rn, 64]
  float* out2 = out1 + (size_t)Rn * HGC_D;        // [Rh, 64]
  const long long total = (long long)(Rn + Rh) * HGC_D;

  // 1) dense GEMMs on the matrix pipe (fp32 WMMA)
  hgc_gemm_xw_wmma<<<(Rn + 15) / 16, 128, 0, stream>>>(node_input, weight, S1, Rn);
  hgc_gemm_xw_wmma<<<(Rh + 15) / 16, 128, 0, stream>>>(he_input, weight, S2, Rh);

  // 2) zero the accumulation buffer (d_out is poisoned by the harness)
  hgc_zero<<<2048, 256, 0, stream>>>((float*)d_out, total);

  // 3) COO SpMM scatter (support tiles live in the 192MB L2)
  {
    long long tn = (long long)nnz_n * 16;
    hgc_spmm_scatter<<<(unsigned)((tn + 255) / 256), 256, 0, stream>>>(
        nrows, ncols, nvals, S1, out1, nnz_n);
    long long th = (long long)nnz_h * 16;
    hgc_spmm_scatter<<<(unsigned)((th + 255) / 256), 256, 0, stream>>>(
        hrows, hcols, hvals, S2, out2, nnz_h);
  }

  // 4) fused relu + bias over the concatenated output
  hgc_relu_bias<<<2048, 256, 0, stream>>>((float*)d_out, bias, total);
}